// LigerGraniteMoeSharedMoESwiGLUMLP_48438641164667
// MI455X (gfx1250) — compile-verified
//
#include <hip/hip_runtime.h>
#include <math.h>

#define T_TOK 4096
#define DM    1024
#define DFF   2048
#define DFF2  4096
#define NE    8

#define XPAD  (DM + 4)    // 1028 floats: +4 pad -> bank shift of 4/row, conflict-free b64
#define APAD  (DFF + 4)   // 2052 floats

typedef __attribute__((ext_vector_type(2))) float v2f;
typedef __attribute__((ext_vector_type(8))) float v8f;

static __device__ __forceinline__ v8f wmma_f32(v2f a, v2f b, v8f c) {
    return __builtin_amdgcn_wmma_f32_16x16x4_f32(false, a, false, b, (short)0, c, false, false);
}

// ---------------- kernel 0: zero the per-expert counters -------------------
__global__ void zero_counts_kernel(int* counts) {
    if (threadIdx.x < NE) counts[threadIdx.x] = 0;
}

// ---------------- kernel 1: router logits + top-2 + gather lists -----------
// one wave (32 lanes) per token
__global__ void router_topk_kernel(const float* __restrict__ x,
                                   const float* __restrict__ rw,
                                   float* __restrict__ logits_out,
                                   int* __restrict__ counts,
                                   int* __restrict__ tlist,
                                   int* __restrict__ klist,
                                   float* __restrict__ glist) {
    const int lane = threadIdx.x & 31;
    const int wave = threadIdx.x >> 5;
    const int t = blockIdx.x * 4 + wave;
    if (t >= T_TOK) return;

    const float* xr = x + (size_t)t * DM;
    float acc[NE];
#pragma unroll
    for (int e = 0; e < NE; ++e) acc[e] = 0.f;

    for (int i = lane; i < DM; i += 32) {
        float xv = xr[i];
#pragma unroll
        for (int e = 0; e < NE; ++e) acc[e] += xv * rw[e * DM + i];
    }
#pragma unroll
    for (int e = 0; e < NE; ++e) {
        float v = acc[e];
#pragma unroll
        for (int off = 16; off > 0; off >>= 1) v += __shfl_xor(v, off, 32);
        acc[e] = v;
    }
    if (lane == 0) {
#pragma unroll
        for (int e = 0; e < NE; ++e) logits_out[t * NE + e] = acc[e];
        // top-2, ties -> lower index (matches lax.top_k)
        int i0 = 0; float v0 = acc[0];
#pragma unroll
        for (int e = 1; e < NE; ++e) if (acc[e] > v0) { v0 = acc[e]; i0 = e; }
        int i1 = (i0 == 0) ? 1 : 0; float v1 = acc[i1];
#pragma unroll
        for (int e = 0; e < NE; ++e) if (e != i0 && acc[e] > v1) { v1 = acc[e]; i1 = e; }
        float ex = __expf(v1 - v0);           // v1 <= v0, safe
        float inv = 1.f / (1.f + ex);
        float g0 = inv;
        float g1 = ex * inv;

        int p0 = atomicAdd(&counts[i0], 1);
        tlist[i0 * T_TOK + p0] = t; klist[i0 * T_TOK + p0] = 0; glist[i0 * T_TOK + p0] = g0;
        int p1 = atomicAdd(&counts[i1], 1);
        tlist[i1 * T_TOK + p1] = t; klist[i1 * T_TOK + p1] = 1; glist[i1 * T_TOK + p1] = g1;
    }
}

// ---------------- kernel 2: fused grouped GEMM1 -> SwiGLU -> GEMM2 ---------
// grid: (1, 256, NE); block: 512 threads = 16 waves; ~197 KB dynamic LDS
__global__ void moe_expert_kernel(const float* __restrict__ x,
                                  const float* __restrict__ w_in,
                                  const float* __restrict__ w_out,
                                  const int* __restrict__ counts,
                                  const int* __restrict__ tlist,
                                  const int* __restrict__ klist,
                                  const float* __restrict__ glist,
                                  float* __restrict__ slotY) {
    extern __shared__ float smem[];
    float* xs = smem;                       // 16 x XPAD  (gathered x tile)
    float* as = smem + 16 * XPAD;           // 16 x APAD  (SwiGLU activations)
    float* gs = as + 16 * APAD;             // 16 gates
    int* toks = (int*)(gs + 16);            // 16 token ids
    int* ks   = toks + 16;                  // 16 slot ids

    const int e = blockIdx.z;
    const int tile = blockIdx.y;
    const int cnt = counts[e];
    const int row0 = tile * 16;
    if (row0 >= cnt) return;
    const int rows = min(16, cnt - row0);

    const int tid = threadIdx.x;
    if (tid < 16) {
        if (tid < rows) {
            toks[tid] = tlist[e * T_TOK + row0 + tid];
            ks[tid]   = klist[e * T_TOK + row0 + tid];
            gs[tid]   = glist[e * T_TOK + row0 + tid];
        } else { toks[tid] = 0; ks[tid] = 0; gs[tid] = 0.f; }
    }
    __syncthreads();

    // gather 16 token rows of x into LDS (zero-pad missing rows)
    for (int i = tid; i < 16 * DM; i += 512) {
        int r = i >> 10;          // / DM
        int c = i & (DM - 1);
        float v = (r < rows) ? x[(size_t)toks[r] * DM + c] : 0.f;
        xs[r * XPAD + c] = v;
    }
    __syncthreads();

    const int lane = tid & 31;
    const int wave = tid >> 5;        // 0..15
    const int half = lane >> 4;       // 0/1
    const int ln   = lane & 15;

    // ---- GEMM1 (+SwiGLU): h = x @ w_in[e]^T, cols split h1|h2 ----
    const float* wi = w_in + (size_t)e * DFF2 * DM;
    const float* apx = xs + ln * XPAD + 2 * half;
#pragma unroll 1
    for (int p = 0; p < 8; ++p) {
        const int fb = wave * 128 + p * 16;                 // h1 column base (0..2047)
        const float* b1p = wi + (size_t)(fb + ln) * DM + 2 * half;
        const float* b2p = wi + (size_t)(fb + DFF + ln) * DM + 2 * half;
        v8f c1 = {}; v8f c2 = {};
#pragma unroll 4
        for (int k = 0; k < DM; k += 4) {
            v2f a  = *(const v2f*)(apx + k);
            v2f b1 = *(const v2f*)(b1p + k);
            v2f b2 = *(const v2f*)(b2p + k);
            c1 = wmma_f32(a, b1, c1);
            c2 = wmma_f32(a, b2, c2);
        }
#pragma unroll
        for (int j = 0; j < 8; ++j) {
            int m = j + 8 * half;
            float h1 = c1[j], h2 = c2[j];
            // fast silu: v_exp_f32 + v_rcp_f32, no IEEE div sequence
            float sig = __builtin_amdgcn_rcpf(1.f + __expf(-h1));
            as[m * APAD + fb + ln] = h1 * sig * h2;
        }
    }
    __syncthreads();

    // ---- GEMM2: y = act @ w_out[e]^T, scale by gate, scatter to slots ----
    const float* wo = w_out + (size_t)e * DM * DFF;
    const float* apa = as + ln * APAD + 2 * half;
#pragma unroll 1
    for (int pc = 0; pc < 2; ++pc) {
        const int db = wave * 64 + pc * 32;                 // two 16-col tiles
        const float* bp0 = wo + (size_t)(db + ln) * DFF + 2 * half;
        const float* bp1 = wo + (size_t)(db + 16 + ln) * DFF + 2 * half;
        v8f c0 = {}; v8f c1 = {};
#pragma unroll 4
        for (int k = 0; k < DFF; k += 4) {
            v2f a  = *(const v2f*)(apa + k);
            v2f b0 = *(const v2f*)(bp0 + k);
            v2f b1 = *(const v2f*)(bp1 + k);
            c0 = wmma_f32(a, b0, c0);
            c1 = wmma_f32(a, b1, c1);
        }
#pragma unroll
        for (int j = 0; j < 8; ++j) {
            int m = j + 8 * half;
            if (m < rows) {
                float g = gs[m];
                size_t base = ((size_t)toks[m] * 2 + ks[m]) * DM;
                slotY[base + db + ln]      = c0[j] * g;
                slotY[base + db + 16 + ln] = c1[j] * g;
            }
        }
    }
}

// ---------------- kernel 3: deterministic slot combine ---------------------
__global__ void combine_kernel(const float* __restrict__ slotY,
                               float* __restrict__ out) {
    int i = blockIdx.x * blockDim.x + threadIdx.x;
    if (i < T_TOK * DM) {
        int t = i / DM, d = i - t * DM;
        out[i] = slotY[((size_t)t * 2) * DM + d] + slotY[((size_t)t * 2 + 1) * DM + d];
    }
}

extern "C" void kernel_launch(void* const* d_in, const int* in_sizes, int n_in,
                              void* d_out, int out_size, void* d_ws, size_t ws_size,
                              hipStream_t stream) {
    const float* x     = (const float*)d_in[0];   // [2,2048,1024]
    const float* rw    = (const float*)d_in[1];   // [8,1024]
    const float* w_in  = (const float*)d_in[2];   // [8,4096,1024]
    const float* w_out = (const float*)d_in[3];   // [8,1024,2048]

    float* out    = (float*)d_out;                // [T, DM]
    float* logits = out + (size_t)T_TOK * DM;     // [T, NE]

    char* ws = (char*)d_ws;
    int*   counts = (int*)ws;                                  // 8 ints (+pad to 32B)
    int*   tlist  = (int*)(ws + 32);                           // NE*T
    int*   klist  = tlist + (size_t)NE * T_TOK;                // NE*T
    float* glist  = (float*)(klist + (size_t)NE * T_TOK);      // NE*T
    float* slotY  = glist + (size_t)NE * T_TOK;                // T*2*DM

    zero_counts_kernel<<<1, 32, 0, stream>>>(counts);
    router_topk_kernel<<<T_TOK / 4, 128, 0, stream>>>(x, rw, logits, counts,
                                                      tlist, klist, glist);

    size_t smem = (size_t)(16 * XPAD + 16 * APAD + 16) * sizeof(float) + 32 * sizeof(int);
    (void)hipFuncSetAttribute((const void*)moe_expert_kernel,
                              hipFuncAttributeMaxDynamicSharedMemorySize, (int)smem);
    dim3 g2(1, T_TOK / 16, NE);
    moe_expert_kernel<<<g2, 512, smem, stream>>>(x, w_in, w_out, counts,
                                                 tlist, klist, glist, slotY);

    combine_kernel<<<(T_TOK * DM + 255) / 256, 256, 0, stream>>>(slotY, out);
}